// Window_89249420411513
// MI455X (gfx1250) — compile-verified
//
#include <hip/hip_runtime.h>
#include <math.h>

// ---- sizes (compile-time, match the reference) ----
#define BATCH 4096
#define CELL  400
#define KMIX  10
#define TLEN  256
#define VOC   80

typedef __attribute__((ext_vector_type(2))) float v2f;
typedef __attribute__((ext_vector_type(4))) float v4f;
typedef __attribute__((ext_vector_type(8))) float v8f;

// =====================================================================
// Kernel 1: params = x @ W^T + b  via V_WMMA_F32_16X16X4_F32,
// fused epilogue: alpha=exp, beta=exp, kappa=kappa_old+exp (kappa -> d_out).
//
// One wave computes one 16(M) x 32(N) tile (N padded 30->32, two WMMA
// N-tiles). K loop: 100 steps of 4.
//
// Fragment layouts (ISA 7.12.2, wave32):
//   A 16x4 f32 : lanes 0-15 -> M=lane, VGPR{0,1}=K{0,1}; lanes 16-31 -> K{2,3}
//   B 4x16 f32 : lanes 0-15 -> N=lane, VGPR{0,1}=K{0,1}; lanes 16-31 -> K{2,3}
//   C/D 16x16  : VGPR v, lanes 0-15 -> M=v, lanes 16-31 -> M=v+8; N=lane%16
// =====================================================================
__global__ __launch_bounds__(256) void window_gemm_params(
    const float* __restrict__ x,        // [B, CELL]
    const float* __restrict__ Wmat,     // [3K, CELL]
    const float* __restrict__ bias,     // [3K]
    const float* __restrict__ kappa_old,// [B, K]
    float* __restrict__ ws_alpha,       // [B, K]
    float* __restrict__ ws_beta,        // [B, K]
    float* __restrict__ ws_kappa,       // [B, K]
    float* __restrict__ out_kappa)      // [B, K] (second output region)
{
    const int wave  = threadIdx.x >> 5;        // 8 waves / block
    const int lane  = threadIdx.x & 31;
    const int lhalf = lane >> 4;               // 0: K{0,1}, 1: K{2,3}
    const int lmod  = lane & 15;

    const int mtile = blockIdx.x * 8 + wave;   // 32 blocks * 8 waves = 256 tiles
    const int r0    = mtile * 16;              // first row of this M tile

    // A pointer: row r0+lmod, k pairs at k0 + 2*lhalf
    const float* aRow = x + (size_t)(r0 + lmod) * CELL + 2 * lhalf;

    // B pointers for the two N tiles (columns of params = rows of W)
    const int n0 = lmod;            // tile 0 column
    const int n1 = 16 + lmod;       // tile 1 column (>=30 -> zero-fill)
    const float* bRow0 = Wmat + (size_t)n0 * CELL + 2 * lhalf;
    const bool  n1ok  = (n1 < 3 * KMIX);
    const float* bRow1 = Wmat + (size_t)(n1ok ? n1 : 0) * CELL + 2 * lhalf;

    v8f acc0 = {};
    v8f acc1 = {};

#pragma unroll 4
    for (int k0 = 0; k0 < CELL; k0 += 4) {
        v2f a  = *(const v2f*)(aRow  + k0);
        v2f b0 = *(const v2f*)(bRow0 + k0);
        v2f b1 = *(const v2f*)(bRow1 + k0);
        if (!n1ok) { b1.x = 0.f; b1.y = 0.f; }
        acc0 = __builtin_amdgcn_wmma_f32_16x16x4_f32(
            false, a, false, b0, (short)0, acc0, false, false);
        acc1 = __builtin_amdgcn_wmma_f32_16x16x4_f32(
            false, a, false, b1, (short)0, acc1, false, false);
    }

    // Epilogue: this lane owns column n for 8 rows per N tile.
#pragma unroll
    for (int ntile = 0; ntile < 2; ++ntile) {
        const int n = ntile * 16 + lmod;
        if (n >= 3 * KMIX) continue;
        const float bn = bias[n];
#pragma unroll
        for (int v = 0; v < 8; ++v) {
            const int row = r0 + v + 8 * lhalf;
            const float val = (ntile == 0 ? acc0[v] : acc1[v]) + bn;
            const float e = __expf(val);
            if (n < KMIX) {
                ws_alpha[row * KMIX + n] = e;
            } else if (n < 2 * KMIX) {
                ws_beta[row * KMIX + (n - KMIX)] = e;
            } else {
                const int kk = n - 2 * KMIX;
                const float kp = kappa_old[row * KMIX + kk] + e;
                ws_kappa[row * KMIX + kk] = kp;
                out_kappa[row * KMIX + kk] = kp;
            }
        }
    }
}

// =====================================================================
// Kernel 2 (bandwidth-critical): one block per batch element.
//   Phase 1: phi[t] = sum_k alpha_k * exp(-beta_k * (kappa_k - t)^2) -> LDS
//   Phase 2: weight[v] = sum_t phi[t] * onehots[b,t,v]
// onehots streamed once as float4, non-temporal (335 MB > 192 MB L2).
// 320 threads = (20 float4 columns) x (16 t-groups); LDS tree-reduce.
// =====================================================================
__global__ __launch_bounds__(320) void window_phi_einsum(
    const float* __restrict__ onehots,  // [B, T, V]
    const float* __restrict__ ws_alpha, // [B, K]
    const float* __restrict__ ws_beta,  // [B, K]
    const float* __restrict__ ws_kappa, // [B, K]
    float* __restrict__ out_weight)     // [B, V]
{
    const int b   = blockIdx.x;
    const int tid = threadIdx.x;

    __shared__ float s_al[KMIX];
    __shared__ float s_be[KMIX];
    __shared__ float s_ka[KMIX];
    __shared__ float s_phi[TLEN];
    __shared__ v4f   s_part[16 * 20];   // [t-group][v4]

    if (tid < KMIX)                 s_al[tid]            = ws_alpha[b * KMIX + tid];
    else if (tid < 2 * KMIX)        s_be[tid - KMIX]     = ws_beta [b * KMIX + (tid - KMIX)];
    else if (tid < 3 * KMIX)        s_ka[tid - 2 * KMIX] = ws_kappa[b * KMIX + (tid - 2 * KMIX)];
    __syncthreads();

    if (tid < TLEN) {
        const float t = (float)tid;
        float p = 0.f;
#pragma unroll
        for (int k = 0; k < KMIX; ++k) {
            const float d = s_ka[k] - t;
            p += s_al[k] * __expf(-s_be[k] * d * d);
        }
        s_phi[tid] = p;
    }
    __syncthreads();

    const int v4 = tid % 20;        // float4 column (80 floats = 20 x float4)
    const int g  = tid / 20;        // 16 t-groups
    const v4f* oh = (const v4f*)(onehots + (size_t)b * TLEN * VOC);

    v4f acc = {};
    if (g < 16) {
#pragma unroll 4
        for (int step = 0; step < 16; ++step) {
            const int t = step * 16 + g;
            const v4f row = __builtin_nontemporal_load(&oh[t * 20 + v4]);
            acc += s_phi[t] * row;
        }
        s_part[g * 20 + v4] = acc;
    }
    __syncthreads();

    if (tid < 20) {
        v4f s = s_part[tid];
#pragma unroll
        for (int gg = 1; gg < 16; ++gg) s += s_part[gg * 20 + tid];
        ((v4f*)out_weight)[b * 20 + tid] = s;
    }
}

// =====================================================================
// Host launcher
// d_in: 0=x [B,CELL], 1=kappa_old [B,K], 2=onehots [B,T,V],
//       3=W [3K,CELL], 4=b [3K], 5=text_len (scalar, unused: T fixed)
// d_out: weight [B*V] then kappa [B*K] (flat, return order)
// =====================================================================
extern "C" void kernel_launch(void* const* d_in, const int* in_sizes, int n_in,
                              void* d_out, int out_size, void* d_ws, size_t ws_size,
                              hipStream_t stream) {
    const float* x       = (const float*)d_in[0];
    const float* kold    = (const float*)d_in[1];
    const float* onehots = (const float*)d_in[2];
    const float* Wmat    = (const float*)d_in[3];
    const float* bias    = (const float*)d_in[4];

    float* out        = (float*)d_out;
    float* out_weight = out;                       // [B*V]
    float* out_kappa  = out + (size_t)BATCH * VOC; // [B*K]

    float* wsf      = (float*)d_ws;
    float* ws_alpha = wsf;
    float* ws_beta  = wsf + (size_t)BATCH * KMIX;
    float* ws_kappa = wsf + (size_t)2 * BATCH * KMIX;

    // 256 M-tiles of 16 rows; 8 waves/block -> 32 blocks of 256 threads
    window_gemm_params<<<dim3(32), dim3(256), 0, stream>>>(
        x, Wmat, bias, kold, ws_alpha, ws_beta, ws_kappa, out_kappa);

    // one block per batch element, streams 80 KB of onehots each
    window_phi_einsum<<<dim3(BATCH), dim3(320), 0, stream>>>(
        onehots, ws_alpha, ws_beta, ws_kappa, out_weight);
}